// Pe_local_based_attention_variation_27367531610178
// MI455X (gfx1250) — compile-verified
//
#include <hip/hip_runtime.h>

#define BB 8
#define HH 8
#define NN 4096
#define KK 32
#define DD 16
#define PSTR 132   // padded LDS row stride (floats) to spread banks

typedef float v2f __attribute__((ext_vector_type(2)));
typedef float v8f __attribute__((ext_vector_type(8)));

__global__ __launch_bounds__(64) void pe_attn_kernel(
    const float* __restrict__ q, const float* __restrict__ kmat,
    const float* __restrict__ v, const float* __restrict__ xyz,
    const float* __restrict__ W1, const float* __restrict__ b1,
    const float* __restrict__ W2, const float* __restrict__ b2,
    float* __restrict__ out)
{
  __shared__ float Plds[2][32 * PSTR];   // per-wave 32x128 projection

  const int tid  = threadIdx.x;
  const int wave = tid >> 5;
  const int lane = tid & 31;
  const int gidx = blockIdx.x * 2 + wave;   // 8-position group id, 0..32767
  const int nblk = gidx & 511;              // n >> 3
  const int h    = (gidx >> 9) & 7;
  const int b    = gidx >> 12;
  const int n0     = nblk * 8;
  const int nprime = h * 512 + nblk;        // scrambled-reshape source row

  const bool hi = lane >= 16;
  const int  m  = lane & 15;

  // ---------- WMMA: P = xyz[b, nprime, 0..31, :] @ W2 + b2  (32 x 128) ----------
  // A (16x4 f32): VGPR0 = K0 (lanes 0-15) / K2 (lanes 16-31); VGPR1 = K1 / K3.
  // Branch-free operand build: load all channels unconditionally, select with
  // v_cndmask so EXEC stays all-ones (no saveexec/branch around loads).
  const float* xp = xyz + ((long)(b * NN + nprime)) * (KK * 3);
  const float a00 = xp[m * 3 + 0];
  const float a01 = xp[m * 3 + 1];
  const float a02 = xp[m * 3 + 2];
  const float a10 = xp[(m + 16) * 3 + 0];
  const float a11 = xp[(m + 16) * 3 + 1];
  const float a12 = xp[(m + 16) * 3 + 2];
  v2f A0, A1;
  A0.x = hi ? a02 : a00;
  A0.y = hi ? 0.0f : a01;
  A1.x = hi ? a12 : a10;
  A1.y = hi ? 0.0f : a11;

  float* Pw = &Plds[wave][0];
  #pragma unroll
  for (int nt = 0; nt < 8; ++nt) {
    const int o = nt * 16 + m;              // output column 0..127
    const float w2r0 = W2[o];               // unconditional row loads
    const float w2r1 = W2[128 + o];
    const float w2r2 = W2[256 + o];
    v2f Bv;                                 // B (4x16): rows of W2 (row 3 = 0)
    Bv.x = hi ? w2r2 : w2r0;
    Bv.y = hi ? 0.0f : w2r1;
    const float bias = b2[o];               // C depends only on column
    v8f C = { bias, bias, bias, bias, bias, bias, bias, bias };
    v8f D0 = __builtin_amdgcn_wmma_f32_16x16x4_f32(false, A0, false, Bv,
                                                   (short)0, C, false, false);
    v8f D1 = __builtin_amdgcn_wmma_f32_16x16x4_f32(false, A1, false, Bv,
                                                   (short)0, C, false, false);
    const int rb = hi ? 8 : 0;              // D: VGPR r -> row r (+8 for hi half)
    #pragma unroll
    for (int r = 0; r < 8; ++r) {
      Pw[(rb + r) * PSTR + o]      = D0[r];
      Pw[(16 + rb + r) * PSTR + o] = D1[r];
    }
  }
  __syncthreads();

  // wave-uniform W1 column + bias for e_xyz
  const float w10 = W1[0 * HH + h];
  const float w11 = W1[1 * HH + h];
  const float w12 = W1[2 * HH + h];
  const float b1h = b1[h];

  const int half = lane >> 4;
  const int dd   = lane & 15;

  for (int p = 0; p < 8; ++p) {
    const int n = n0 + p;
    const long bhn = (long)((b * HH + h) * NN + n);
    const float* qP = q    + bhn * DD;
    const float* kP = kmat + bhn * (DD * KK);
    const float* vP = v    + bhn * (KK * DD);
    const float* xe = xyz + ((long)(b * NN + n)) * (KK * 3);

    // prefetch next position's k/v streams (speculative, whole 2KB tiles)
    __builtin_prefetch(kP + DD * KK + lane * 16, 0, 0);
    __builtin_prefetch(vP + KK * DD + lane * 16, 0, 0);

    // ---- energy for neighbor k = lane: q·k[:,lane] + e_xyz ----
    float acc = 0.0f;
    #pragma unroll
    for (int d = 0; d < DD; ++d)
      acc = fmaf(qP[d], kP[d * KK + lane], acc);   // coalesced b32 across lanes
    const float e = fmaf(xe[lane * 3 + 0], w10,
                    fmaf(xe[lane * 3 + 1], w11,
                    fmaf(xe[lane * 3 + 2], w12, b1h)));
    const float logit = (acc + e) * 0.25f;         // / sqrt(16)

    // ---- softmax across the 32 lanes ----
    float mx = logit;
    #pragma unroll
    for (int s = 16; s > 0; s >>= 1)
      mx = fmaxf(mx, __shfl_xor(mx, s, 32));
    const float ex = __expf(logit - mx);
    float sum = ex;
    #pragma unroll
    for (int s = 16; s > 0; s >>= 1)
      sum += __shfl_xor(sum, s, 32);
    const float a = ex / sum;

    // ---- out[d] = sum_k a[k] * (v[k,d] + P[p*4 + k>>3, (k&7)*16 + d]) ----
    const float* Pp = Pw + (p * 4) * PSTR;
    float accv = 0.0f;
    #pragma unroll
    for (int j = 0; j < 16; ++j) {
      const int kk   = (half << 4) + j;
      const float aj = __shfl(a, kk, 32);
      const float vv = vP[kk * DD + dd];           // two 64B segments / iter
      const float pe = Pp[(kk >> 3) * PSTR + (kk & 7) * DD + dd];
      accv = fmaf(aj, vv + pe, accv);
    }
    accv += __shfl_xor(accv, 16, 32);              // combine the two k-halves
    if (lane < 16)                                  // (B,N,H,D) transposed store
      out[(((long)(b * NN + n)) * HH + h) * DD + dd] = accv;
  }
}

extern "C" void kernel_launch(void* const* d_in, const int* in_sizes, int n_in,
                              void* d_out, int out_size, void* d_ws, size_t ws_size,
                              hipStream_t stream) {
  const float* q   = (const float*)d_in[0];
  const float* k   = (const float*)d_in[1];
  const float* v   = (const float*)d_in[2];
  const float* xyz = (const float*)d_in[3];
  const float* W1  = (const float*)d_in[4];
  const float* b1  = (const float*)d_in[5];
  const float* W2  = (const float*)d_in[6];
  const float* b2  = (const float*)d_in[7];
  float* out = (float*)d_out;

  dim3 grid(BB * HH * (NN / 8) / 2);   // 16384 blocks, 2 waves each
  dim3 block(64);
  hipLaunchKernelGGL(pe_attn_kernel, grid, block, 0, stream,
                     q, k, v, xyz, W1, b1, W2, b2, out);
}